// SparseVoxelEncoder_15401752723821
// MI455X (gfx1250) — compile-verified
//
#include <hip/hip_runtime.h>

// SparseVoxelEncoder trilinear embedding interpolation (NSVF style), MI455X / gfx1250.
//
// wave32 mapping: 4 points/wave, 8 lanes/point, each lane owns one float4 of the
// 32-dim embedding. Per point: 8 gathered global_load_b128 from `values`
// (15.4MB -> resident in 192MB L2), 32 FMAs, one coalesced NT b128 store.
// All table offsets fit in 32 bits -> saddr + u32-voffset addressing.
//
// CDNA5-specific paths used:
//  * global_load_async_to_lds_b128/b32 (ASYNCcnt) stages per-point metadata
//    (12 contiguous xyz dwords + 4 sidx dwords per wave) into a wave-private
//    LDS slice; synced with s_wait_asynccnt 0 (no workgroup barrier needed).
//    readfirstlane(pw) makes the common/tail split a true scalar branch:
//    common path = 4 async b128 issues/wave off SGPR bases.
//  * TH temporal hints: NT store for the 128MB write-once output.
// WMMA is mathematically inapplicable: einsum('pk,pkd->pd') has a per-point
// B matrix, so there is no shared-operand matmul to map onto v_wmma.

#define VOXEL_INV 4.0f   // 1 / VOXEL_SIZE (0.25)

typedef float  f32x4 __attribute__((ext_vector_type(4)));
typedef int    i32x4 __attribute__((ext_vector_type(4)));

__global__ __launch_bounds__(256, 8)
void sve_trilinear_kernel(const float* __restrict__ xyz,      // [P,3]
                          const float* __restrict__ centers,  // [V,3]
                          const float* __restrict__ values,   // [K,32]
                          const int*   __restrict__ sidx,     // [P]
                          const int*   __restrict__ feats,    // [V,8]
                          float*       __restrict__ out,      // [P,32]
                          int P)
{
    // Per-wave LDS slice: [0..11] = xyz dwords of the wave's 4 points
    //                     (contiguous in memory: xyz[3*pw .. 3*pw+11]),
    //                     [12..15] = sampled_idx of the 4 points.
    __shared__ int meta[8][16];

    const int tid  = threadIdx.x;
    const int w    = tid >> 5;                  // wave index within block
    const int lane = tid & 31;
    // First point of this wave; identical across the wave -> pin it to an SGPR
    // so path selection is a scalar branch and bases are SALU-computed.
    const int pws  = __builtin_amdgcn_readfirstlane((blockIdx.x * 8 + w) * 4);

    // ---- Stage 1: async-DMA metadata into LDS (gfx1250 ASYNCcnt path) ----
    if (pws + 4 <= P) {
        // Common case (scalar branch): 4 async b128 loads per wave.
        // 12*pws bytes and 4*pws bytes are both 16B-aligned (pws % 4 == 0).
        if (lane < 4) {
            const char* xb = (const char*)(xyz + 3 * pws);   // SGPR base
            const char* sb = (const char*)(sidx + pws);      // SGPR base
            const char* gsrc = (lane < 3) ? (xb + lane * 16) : sb;
            const unsigned ldsoff = (unsigned)(size_t)&meta[w][lane * 4];
            asm volatile("global_load_async_to_lds_b128 %0, %1, off"
                         :: "v"(ldsoff),
                            "v"((unsigned long long)(size_t)gsrc)
                         : "memory");
        }
    } else {
        // Tail wave only: per-dword b32 loads with clamped addresses; clamped
        // dwords land in slots whose consumer lanes exit early.
        if (lane < 16) {
            const int xoff = 3 * pws + lane;                // xyz dword index
            const int xcl  = xoff < 3 * P ? xoff : (3 * P - 1);
            const int soff = pws + (lane - 12);             // sidx index
            const int scl  = soff < P ? soff : (P - 1);
            const void* gsrc = (lane < 12)
                ? (const void*)(xyz + xcl)
                : (const void*)(sidx + scl);
            const unsigned ldsoff = (unsigned)(size_t)&meta[w][lane];
            asm volatile("global_load_async_to_lds_b32 %0, %1, off"
                         :: "v"(ldsoff),
                            "v"((unsigned long long)(size_t)gsrc)
                         : "memory");
        }
    }
    asm volatile("s_wait_asynccnt 0x0" ::: "memory");

    // ---- Stage 2: per-lane trilinear gather/accumulate ----
    const int g8   = lane >> 3;   // point-in-wave (0..3)
    const int quad = lane & 7;    // which float4 of the 32-dim embedding
    const int p    = pws + g8;
    if (p >= P) return;

    const int      b3 = 3 * g8;   // shift+add, no div/mod anywhere
    const unsigned v  = (unsigned)meta[w][12 + g8];
    const float    sx = __int_as_float(meta[w][b3 + 0]);
    const float    sy = __int_as_float(meta[w][b3 + 1]);
    const float    sz = __int_as_float(meta[w][b3 + 2]);

    // Hot tables (reused ~10x per voxel, L2/WGP$-resident): default RT policy.
    const float* crow = (const float*)((const char*)centers + v * 12u);
    const float cx = crow[0];
    const float cy = crow[1];
    const float cz = crow[2];

    // All 8 vertex indices up front (two b128 loads) so the 8 value-row
    // gathers below issue back-to-back (8 loads in flight).
    const char* frow = (const char*)feats + (v << 5);       // v * 8 * 4B
    const i32x4 f0 = *(const i32x4*)(frow);
    const i32x4 f1 = *(const i32x4*)(frow + 16);

    // Relative position inside the voxel, in [0,1].
    const float px = (sx - cx) * VOXEL_INV + 0.5f;
    const float py = (sy - cy) * VOXEL_INV + 0.5f;
    const float pz = (sz - cz) * VOXEL_INV + 0.5f;

    // Trilinear weights; corner k has bits q=((k>>2)&1,(k>>1)&1,k&1) (x-major).
    const float wx0 = 1.0f - px, wx1 = px;
    const float wy0 = 1.0f - py, wy1 = py;
    const float wz0 = 1.0f - pz, wz1 = pz;
    const float a00 = wx0 * wy0, a01 = wx0 * wy1;
    const float a10 = wx1 * wy0, a11 = wx1 * wy1;
    const float w0 = a00 * wz0, w1 = a00 * wz1;
    const float w2 = a01 * wz0, w3 = a01 * wz1;
    const float w4 = a10 * wz0, w5 = a10 * wz1;
    const float w6 = a11 * wz0, w7 = a11 * wz1;

    // Gather this lane's float4 slice of each of the 8 vertex embedding rows.
    // Row byte offset = vert*128 + quad*16 (< 16MB, u32): saddr + voffset form.
    const char*    vbase = (const char*)values;
    const unsigned qb    = (unsigned)quad * 16u;
    const f32x4 e0 = *(const f32x4*)(vbase + (((unsigned)f0.x << 7) + qb));
    const f32x4 e1 = *(const f32x4*)(vbase + (((unsigned)f0.y << 7) + qb));
    const f32x4 e2 = *(const f32x4*)(vbase + (((unsigned)f0.z << 7) + qb));
    const f32x4 e3 = *(const f32x4*)(vbase + (((unsigned)f0.w << 7) + qb));
    const f32x4 e4 = *(const f32x4*)(vbase + (((unsigned)f1.x << 7) + qb));
    const f32x4 e5 = *(const f32x4*)(vbase + (((unsigned)f1.y << 7) + qb));
    const f32x4 e6 = *(const f32x4*)(vbase + (((unsigned)f1.z << 7) + qb));
    const f32x4 e7 = *(const f32x4*)(vbase + (((unsigned)f1.w << 7) + qb));

    f32x4 acc = w0 * e0;
    acc += w1 * e1;
    acc += w2 * e2;
    acc += w3 * e3;
    acc += w4 * e4;
    acc += w5 * e5;
    acc += w6 * e6;
    acc += w7 * e7;

    // 128MB written once, never re-read -> non-temporal store preserves L2 for tables.
    f32x4* dst = (f32x4*)((char*)out + (((unsigned)p << 7) + qb));
    __builtin_nontemporal_store(acc, dst);
}

extern "C" void kernel_launch(void* const* d_in, const int* in_sizes, int n_in,
                              void* d_out, int out_size, void* d_ws, size_t ws_size,
                              hipStream_t stream) {
    // setup_inputs() order: sampled_xyz, points, values, sampled_idx, feats
    const float* xyz     = (const float*)d_in[0];
    const float* centers = (const float*)d_in[1];
    const float* values  = (const float*)d_in[2];
    const int*   sidx    = (const int*)d_in[3];
    const int*   feats   = (const int*)d_in[4];
    float*       out     = (float*)d_out;

    const int P = in_sizes[3];                   // NUM_POINTS
    const int block = 256;                       // 8 wave32, 32 points per block
    const int grid  = (P + 31) / 32;

    sve_trilinear_kernel<<<grid, block, 0, stream>>>(xyz, centers, values, sidx, feats, out, P);
}